// PSRoIPool_32633161515777
// MI455X (gfx1250) — compile-verified
//
#include <hip/hip_runtime.h>

// PS-RoI average pooling (R-FCN semantics), direct bin summation.
// features: (B=4, C=1029, H=64, W=64) f32, rois: (R, 5) f32
// out: (R, 21, 7, 7) f32
//
// Bin windows are <= ~3x3 elements (wh<=216 px * 0.0625 scale / 7 bins), so
// direct summation beats the reference's summed-area-table (two full 67MB
// passes). The 67MB feature tensor is fully L2-resident on MI455X (192MB L2),
// so the per-channel scattered gathers are L2 hits after first touch.
// Stores are coalesced (pw fastest-varying). No matrix structure exists in
// this op (every output contracts a distinct channel over a data-dependent
// window), so WMMA is not applicable; this is a pure gather kernel.
//
// Boundary math uses __fmul_rn/__fadd_rn to forbid FMA contraction so the
// floor/ceil bin boundaries match the reference's separate mul+add exactly.

#define PSR_PH 7
#define PSR_PW 7
#define PSR_OD 21
#define PSR_G 7
#define PSR_SCALE 0.0625f
#define PSR_B 4
#define PSR_H 64
#define PSR_W 64
#define PSR_C (PSR_OD * PSR_G * PSR_G) // 1029

__global__ __launch_bounds__(256) void psroi_pool_kernel(
    const float* __restrict__ features,
    const float* __restrict__ rois,
    float* __restrict__ out,
    int R)
{
    const int idx = blockIdx.x * 256 + threadIdx.x;
    const int total = R * PSR_OD * PSR_PH * PSR_PW;
    if (idx >= total) return;

    // idx = ((r*OD + od)*PH + ph)*PW + pw  (pw fastest -> coalesced stores)
    const int pw = idx % PSR_PW;
    int t = idx / PSR_PW;
    const int ph = t % PSR_PH;
    t /= PSR_PH;
    const int od = t % PSR_OD;
    const int r  = t / PSR_OD;

    const float* roi = rois + (size_t)r * 5;
    // gfx1250 global_prefetch_b8: warm the ROI descriptor row
    __builtin_prefetch(roi, 0, 3);

    const int   b  = (int)roi[0];
    // jnp.round == round-half-to-even == rintf under default RNE mode
    const float sw = __fmul_rn(rintf(roi[1]),                 PSR_SCALE);
    const float sh = __fmul_rn(rintf(roi[2]),                 PSR_SCALE);
    const float ew = __fmul_rn(rintf(__fadd_rn(roi[3], 1.0f)), PSR_SCALE);
    const float eh = __fmul_rn(rintf(__fadd_rn(roi[4], 1.0f)), PSR_SCALE);

    const float roi_h = fmaxf(__fadd_rn(eh, -sh), 0.1f);
    const float roi_w = fmaxf(__fadd_rn(ew, -sw), 0.1f);
    // true division by 7.0f to keep floor/ceil boundaries bit-identical
    const float bin_h = __fdiv_rn(roi_h, 7.0f);
    const float bin_w = __fdiv_rn(roi_w, 7.0f);

    // clip(floor/ceil(mul + add), 0, H) then cast, exactly as the reference;
    // __fmul_rn/__fadd_rn block FMA contraction at the bin boundaries.
    const int hs = (int)fminf(fmaxf(floorf(__fadd_rn(__fmul_rn((float)ph,       bin_h), sh)), 0.0f), (float)PSR_H);
    const int he = (int)fminf(fmaxf(ceilf (__fadd_rn(__fmul_rn((float)(ph + 1), bin_h), sh)), 0.0f), (float)PSR_H);
    const int ws = (int)fminf(fmaxf(floorf(__fadd_rn(__fmul_rn((float)pw,       bin_w), sw)), 0.0f), (float)PSR_W);
    const int we = (int)fminf(fmaxf(ceilf (__fadd_rn(__fmul_rn((float)(pw + 1), bin_w), sw)), 0.0f), (float)PSR_W);

    // position-sensitive channel: G == PH == PW == 7 -> gh==ph, gw==pw
    const int c = (od * PSR_G + ph) * PSR_G + pw;
    const float* plane = features + (size_t)(b * PSR_C + c) * (PSR_H * PSR_W);

    float sum = 0.0f;
    for (int h = hs; h < he; ++h) {
        const float* rowp = plane + h * PSR_W;
        #pragma unroll 4
        for (int w = ws; w < we; ++w) {
            sum += rowp[w];
        }
    }

    const int area = (he - hs) * (we - ws);
    out[idx] = (area > 0) ? __fdiv_rn(sum, (float)area) : 0.0f;
}

extern "C" void kernel_launch(void* const* d_in, const int* in_sizes, int n_in,
                              void* d_out, int out_size, void* d_ws, size_t ws_size,
                              hipStream_t stream) {
    (void)n_in; (void)d_ws; (void)ws_size; (void)out_size;

    const float* features = (const float*)d_in[0];
    const float* rois     = (const float*)d_in[1];
    float*       out      = (float*)d_out;

    const int R     = in_sizes[1] / 5;              // rois: (R, 5)
    const int total = R * PSR_OD * PSR_PH * PSR_PW; // == out_size
    const int blocks = (total + 255) / 256;

    psroi_pool_kernel<<<blocks, 256, 0, stream>>>(features, rois, out, R);
}